// PhysicsAPPNPPropagation_37349035606694
// MI455X (gfx1250) — compile-verified
//
#include <hip/hip_runtime.h>
#include <hip/hip_bf16.h>

#define D_FEAT   48
#define KITER    10
#define ALPHA    0.1f
#define EDGE_CAP 2048   // per-block LDS edge cap (16 KB of int2); avg need ~512

// ---------------------------------------------------------------------------
// Kernel 1: zero the per-node accumulators (deg_w float sums, cnt int counts)
// ---------------------------------------------------------------------------
__global__ void appnp_zero_kernel(float* __restrict__ deg_w,
                                  int* __restrict__ cnt, int n) {
    int i = blockIdx.x * blockDim.x + threadIdx.x;
    if (i < n) { deg_w[i] = 0.0f; cnt[i] = 0; }
}

// ---------------------------------------------------------------------------
// Kernel 2: per-edge degree count + weighted degree sum (row = destination)
// ---------------------------------------------------------------------------
__global__ void appnp_count_kernel(const int* __restrict__ row,
                                   const float* __restrict__ w,
                                   float* __restrict__ deg_w,
                                   int* __restrict__ cnt, int E) {
    int e = blockIdx.x * blockDim.x + threadIdx.x;
    if (e < E) {
        int r = row[e];
        atomicAdd(&cnt[r], 1);
        atomicAdd(&deg_w[r], w[e]);
    }
}

// ---------------------------------------------------------------------------
// Kernel 3: single-block tiled exclusive scan -> rowptr[N+1], cursor[N]
// ---------------------------------------------------------------------------
__global__ __launch_bounds__(1024)
void appnp_scan_kernel(const int* __restrict__ cnt,
                       int* __restrict__ rowptr,
                       int* __restrict__ cursor, int n) {
    __shared__ int smem[1024];
    __shared__ int s_carry;
    if (threadIdx.x == 0) s_carry = 0;
    __syncthreads();
    for (int base = 0; base < n; base += 1024) {
        int i = base + (int)threadIdx.x;
        int v = (i < n) ? cnt[i] : 0;
        smem[threadIdx.x] = v;
        __syncthreads();
        for (int off = 1; off < 1024; off <<= 1) {
            int t = (threadIdx.x >= (unsigned)off) ? smem[threadIdx.x - off] : 0;
            __syncthreads();
            smem[threadIdx.x] += t;
            __syncthreads();
        }
        int excl = smem[threadIdx.x] - v;
        if (i < n) {
            int p = s_carry + excl;
            rowptr[i] = p;
            cursor[i] = p;
        }
        __syncthreads();
        if (threadIdx.x == 1023) s_carry += smem[1023];
        __syncthreads();
    }
    if (threadIdx.x == 0) rowptr[n] = s_carry;   // == E
}

// ---------------------------------------------------------------------------
// Kernel 4: scatter edges into CSR slots as packed {col, norm} records
// ---------------------------------------------------------------------------
__global__ void appnp_scatter_kernel(const int* __restrict__ row,
                                     const int* __restrict__ col,
                                     const float* __restrict__ w,
                                     const float* __restrict__ deg_w,
                                     int* __restrict__ cursor,
                                     int2* __restrict__ edges, int E) {
    int e = blockIdx.x * blockDim.x + threadIdx.x;
    if (e < E) {
        int r = row[e];
        float nrm = w[e] / (deg_w[r] + 1e-10f);
        int pos = atomicAdd(&cursor[r], 1);
        edges[pos] = make_int2(col[e], __float_as_int(nrm));
    }
}

// ---------------------------------------------------------------------------
// Kernel 5: one propagation step, gather-only (no atomics).
// Block = (12, 32): 12 lanes per node, each lane owns a float4 (48 = 12*4).
// The block's contiguous CSR edge segment is async-DMA'd into LDS
// (global_load_async_to_lds_b64, waited with s_wait_asynccnt), so the VMEM
// pipe is reserved for the L2-resident 16B h-gathers while edge records are
// served by conflict-free LDS broadcasts.
// ---------------------------------------------------------------------------
__global__ __launch_bounds__(384)
void appnp_prop_kernel(const float* __restrict__ h_in,
                       const float* __restrict__ x,
                       const int* __restrict__ rowptr,
                       const int2* __restrict__ edges,
                       float* __restrict__ out, int N) {
    __shared__ int2 sEdges[EDGE_CAP];

    const int node0 = blockIdx.x * 32;
    const int node  = node0 + threadIdx.y;
    const int t     = threadIdx.x;                       // 0..11, float4 slot
    const int tid   = threadIdx.x + threadIdx.y * 12;    // 0..383

    const int lastNode    = min(node0 + 32, N);
    const int blockStart  = rowptr[node0];
    const int blockEnd    = rowptr[lastNode];
    const int nE          = blockEnd - blockStart;

    const bool useLds = (nE <= EDGE_CAP);                // block-uniform
    if (useLds) {
        unsigned long long gbase =
            (unsigned long long)(const void*)(edges + blockStart);
        for (int i = tid; i < nE; i += 384) {
            unsigned lds_off = (unsigned)(unsigned long long)(const void*)&sEdges[i];
            unsigned goff    = (unsigned)(i << 3);       // byte offset
            asm volatile("global_load_async_to_lds_b64 %0, %1, %2"
                         :: "v"(lds_off), "v"(goff), "s"(gbase)
                         : "memory");
        }
        asm volatile("s_wait_asynccnt 0x0" ::: "memory");
        __syncthreads();
    }

    if (node >= N) return;
    const int start = rowptr[node];
    const int end   = rowptr[node + 1];

    float4 acc = make_float4(0.0f, 0.0f, 0.0f, 0.0f);

    if (useLds) {
        for (int e = start; e < end; ++e) {
            int2 rec  = sEdges[e - blockStart];          // LDS broadcast
            float nrm = __int_as_float(rec.y);
            const float4* hp = (const float4*)(h_in + (size_t)rec.x * D_FEAT);
            float4 v = hp[t];
            acc.x = fmaf(nrm, v.x, acc.x);
            acc.y = fmaf(nrm, v.y, acc.y);
            acc.z = fmaf(nrm, v.z, acc.z);
            acc.w = fmaf(nrm, v.w, acc.w);
        }
    } else {
        __builtin_prefetch(&edges[start], 0, 0);         // global_prefetch_b8
        for (int e = start; e < end; ++e) {
            int2 rec  = edges[e];
            float nrm = __int_as_float(rec.y);
            const float4* hp = (const float4*)(h_in + (size_t)rec.x * D_FEAT);
            float4 v = hp[t];
            acc.x = fmaf(nrm, v.x, acc.x);
            acc.y = fmaf(nrm, v.y, acc.y);
            acc.z = fmaf(nrm, v.z, acc.z);
            acc.w = fmaf(nrm, v.w, acc.w);
        }
    }

    const float4* xp = (const float4*)(x + (size_t)node * D_FEAT);
    float4 xv = xp[t];
    float4 r;
    r.x = (1.0f - ALPHA) * acc.x + ALPHA * xv.x;
    r.y = (1.0f - ALPHA) * acc.y + ALPHA * xv.y;
    r.z = (1.0f - ALPHA) * acc.z + ALPHA * xv.z;
    r.w = (1.0f - ALPHA) * acc.w + ALPHA * xv.w;
    ((float4*)(out + (size_t)node * D_FEAT))[t] = r;
}

// ---------------------------------------------------------------------------
// Host-side launch
// ---------------------------------------------------------------------------
static inline size_t align256(size_t v) { return (v + 255) & ~(size_t)255; }

extern "C" void kernel_launch(void* const* d_in, const int* in_sizes, int n_in,
                              void* d_out, int out_size, void* d_ws, size_t ws_size,
                              hipStream_t stream) {
    const float* x  = (const float*)d_in[0];           // [N, 48]
    const int*   ei = (const int*)d_in[1];             // [2, E] flattened
    const float* w  = (const float*)d_in[2];           // [E]

    const int N = in_sizes[0] / D_FEAT;
    const int E = in_sizes[1] / 2;
    const int* row = ei;          // edge_index[0] : scatter destination
    const int* col = ei + E;      // edge_index[1] : gather source

    // carve workspace
    char* p = (char*)d_ws;
    float* deg_w  = (float*)p;  p += align256((size_t)N * sizeof(float));
    int*   cnt    = (int*)p;    p += align256((size_t)N * sizeof(int));
    int*   rowptr = (int*)p;    p += align256((size_t)(N + 1) * sizeof(int));
    int*   cursor = (int*)p;    p += align256((size_t)N * sizeof(int));
    int2*  edges  = (int2*)p;   p += align256((size_t)E * sizeof(int2));
    float* hbuf   = (float*)p;  p += align256((size_t)N * D_FEAT * sizeof(float));

    const int T = 256;
    const int gN = (N + T - 1) / T;
    const int gE = (E + T - 1) / T;

    appnp_zero_kernel<<<gN, T, 0, stream>>>(deg_w, cnt, N);
    appnp_count_kernel<<<gE, T, 0, stream>>>(row, w, deg_w, cnt, E);
    appnp_scan_kernel<<<1, 1024, 0, stream>>>(cnt, rowptr, cursor, N);
    appnp_scatter_kernel<<<gE, T, 0, stream>>>(row, col, w, deg_w, cursor, edges, E);

    // K propagation steps, ping-pong hbuf <-> d_out (K even => ends in d_out)
    dim3 pblock(12, 32);
    int  pgrid = (N + 31) / 32;
    const float* cur = x;
    for (int k = 0; k < KITER; ++k) {
        float* o = (k & 1) ? (float*)d_out : hbuf;
        appnp_prop_kernel<<<pgrid, pblock, 0, stream>>>(cur, x, rowptr, edges, o, N);
        cur = o;
    }
}